// TRLayer_74019466379906
// MI455X (gfx1250) — compile-verified
//
#include <hip/hip_runtime.h>
#include <math.h>

typedef __attribute__((ext_vector_type(2))) float v2f;
typedef __attribute__((ext_vector_type(8))) float v8f;

#define Bn 128
#define Sn 512
#define Hn 128
#define PROW 516   // 512 + 4 pad: spreads LDS banks for strided row reads

__device__ __forceinline__ float wave_reduce_sum(float v) {
#pragma unroll
    for (int off = 16; off >= 1; off >>= 1)
        v += __shfl_xor(v, off, 32);
    return v;
}

// ---------------------------------------------------------------------------
// sa[b,i] = dot(hidden[b,i,:], w1) ; sb[b,i] = dot(hidden[b,i,:], w2)
// One wave per row; lane loads float4 (32*4 = 128 = H).
// ---------------------------------------------------------------------------
__global__ void scores_kernel(const float* __restrict__ hidden,
                              const float* __restrict__ w1,
                              const float* __restrict__ w2,
                              float* __restrict__ sa,
                              float* __restrict__ sb) {
    const int wave = threadIdx.x >> 5;
    const int lane = threadIdx.x & 31;
    const int row  = blockIdx.x * 8 + wave;   // 8 waves per 256-thread block
    const float4 h = ((const float4*)(hidden + (size_t)row * Hn))[lane];
    const float4 a = ((const float4*)w1)[lane];
    const float4 c = ((const float4*)w2)[lane];
    float da = h.x * a.x + h.y * a.y + h.z * a.z + h.w * a.w;
    float db = h.x * c.x + h.y * c.y + h.z * c.z + h.w * c.w;
    da = wave_reduce_sum(da);
    db = wave_reduce_sum(db);
    if (lane == 0) { sa[row] = da; sb[row] = db; }
}

// ---------------------------------------------------------------------------
// Fused: masked-tanh scores -> (implicit) softmax -> alpha @ hidden
// One workgroup (256 thr = 8 waves) handles 16 rows of one batch element.
// Phase 1: p[r][j] = softmax numerator (0 if both adj entries are 0),
//          rowinv[r] = 1/rowsum  (softmax denominator folded into store).
// Phase 2: 8 waves, wave w owns output columns [16w, 16w+16):
//          C(16x16) += A(16x4,f32) x B(4x16,f32) via v_wmma_f32_16x16x4_f32,
//          K loop over 512 in steps of 4 (128 WMMAs per wave).
// ---------------------------------------------------------------------------
__global__ void aggregate_kernel(const float* __restrict__ hidden,
                                 const float* __restrict__ adj,
                                 const float* __restrict__ sa,
                                 const float* __restrict__ sb,
                                 const float* __restrict__ bptr,
                                 float* __restrict__ out) {
    __shared__ float p[16 * PROW];
    __shared__ float rowinv[16];

    const int b    = blockIdx.y;
    const int i0   = blockIdx.x * 16;
    const int wave = threadIdx.x >> 5;
    const int lane = threadIdx.x & 31;
    const float bias = bptr[0];

    const float* sa_b  = sa + b * Sn;
    const float* sb_b  = sb + b * Sn;
    const float* adj_b = adj + (size_t)b * Sn * Sn;

    // ---- Phase 1: each wave builds 2 rows of exp-weights in LDS ----
#pragma unroll
    for (int rr = 0; rr < 2; ++rr) {
        const int r = wave * 2 + rr;
        const int i = i0 + r;
        const float sai = sa_b[i];
        const float sbi = sb_b[i];
        float sum = 0.f;
        for (int j = lane; j < Sn; j += 32) {
            const float aij = adj_b[(size_t)i * Sn + j];   // coalesced row
            const float aji = adj_b[(size_t)j * Sn + i];   // transposed (L2-resident)
            const float e1 = (aij == 1.0f) ? tanhf(sai + sb_b[j] + bias) : 0.0f;
            const float e2 = (aji == 1.0f) ? tanhf(sa_b[j] + sbi + bias) : 0.0f;
            const float v  = 0.5f * (e1 + e2);
            // masked entries -> weight 0; tanh in [-1,1] so no max-shift needed
            const float w  = (aij + aji == 0.0f) ? 0.0f : __expf(v);
            p[r * PROW + j] = w;
            sum += w;
        }
        sum = wave_reduce_sum(sum);
        if (lane == 0) rowinv[r] = (sum > 0.f) ? (1.0f / sum) : 0.0f;
    }
    __syncthreads();

    // ---- Phase 2: WMMA f32 16x16x4, K = 512 ----
    const int n0 = wave * 16;          // this wave's output column tile
    const int m  = lane & 15;          // A: M row / B,C,D: N column within tile
    const int hi = lane >> 4;          // half-wave selects K pair / M half
    const float* hb = hidden + (size_t)b * Sn * Hn;

    v8f c = {};
#pragma unroll 4
    for (int k = 0; k < Sn; k += 4) {
        const int kk = k + 2 * hi;     // lanes 0-15: K={k,k+1}; 16-31: K={k+2,k+3}
        v2f a;                         // A fragment from LDS (16x4 tile of p)
        a.x = p[m * PROW + kk];
        a.y = p[m * PROW + kk + 1];
        v2f bv;                        // B fragment from global (4x16 tile of hidden)
        bv.x = hb[(size_t)kk * Hn + n0 + m];
        bv.y = hb[(size_t)(kk + 1) * Hn + n0 + m];
        // Speculative stream-ahead (global_prefetch_b8); invalid addresses are
        // silently dropped per ISA, so no guard -> branch-free inner loop.
        __builtin_prefetch(&hb[(size_t)(k + 8) * Hn + n0 + m], 0, 3);
        c = __builtin_amdgcn_wmma_f32_16x16x4_f32(
                /*neg_a=*/false, a, /*neg_b=*/false, bv,
                /*c_mod=*/(short)0, c, /*reuse_a=*/false, /*reuse_b=*/false);
    }

    // ---- store: fold softmax denominator into each accumulator row ----
#pragma unroll
    for (int v = 0; v < 8; ++v) {
        const int row = v + 8 * hi;    // C/D layout: VGPR v, half-wave hi -> M
        out[((size_t)b * Sn + i0 + row) * Hn + n0 + m] = c[v] * rowinv[row];
    }
}

// ---------------------------------------------------------------------------
// out[b,s,:] = h[b, alias[b,s], :]
// ---------------------------------------------------------------------------
__global__ void gather_kernel(const float* __restrict__ h,
                              const int* __restrict__ alias,
                              float* __restrict__ out) {
    const int b = blockIdx.y;
    const int s = blockIdx.x;
    const int t = threadIdx.x;
    const int src = alias[b * Sn + s];
    out[((size_t)b * Sn + s) * Hn + t] = h[((size_t)b * Sn + src) * Hn + t];
}

extern "C" void kernel_launch(void* const* d_in, const int* in_sizes, int n_in,
                              void* d_out, int out_size, void* d_ws, size_t ws_size,
                              hipStream_t stream) {
    const float* hidden = (const float*)d_in[0];
    const float* adj    = (const float*)d_in[1];
    const int*   alias  = (const int*)d_in[2];
    const float* w1     = (const float*)d_in[3];
    const float* w2     = (const float*)d_in[4];
    const float* bptr   = (const float*)d_in[5];
    float* out = (float*)d_out;

    float* sa = (float*)d_ws;                        // B*S
    float* sb = sa + Bn * Sn;                        // B*S
    float* h1 = sb + Bn * Sn;                        // B*S*H
    float* h2 = h1 + (size_t)Bn * Sn * Hn;           // B*S*H   (~64.5 MB total)

    const dim3 blkS(256), grdS((Bn * Sn) / 8);       // 8 rows per block
    const dim3 blkA(256), grdA(Sn / 16, Bn);         // 16 rows per block, per batch

    // layer 1
    scores_kernel<<<grdS, blkS, 0, stream>>>(hidden, w1, w2, sa, sb);
    aggregate_kernel<<<grdA, blkA, 0, stream>>>(hidden, adj, sa, sb, bptr, h1);
    // layer 2
    scores_kernel<<<grdS, blkS, 0, stream>>>(h1, w1, w2, sa, sb);
    aggregate_kernel<<<grdA, blkA, 0, stream>>>(h1, adj, sa, sb, bptr, h2);
    // gather
    gather_kernel<<<dim3(Sn, Bn), dim3(Hn), 0, stream>>>(h2, alias, out);
}